// ForwardWarpStereo_1133871366641
// MI455X (gfx1250) — compile-verified
//
#include <hip/hip_runtime.h>
#include <float.h>
#include <stdint.h>

// Problem shape (fixed by setup_inputs): im [8,3,720,1280] f32, disp [8,1,720,1280] f32
#define BB 8
#define CC 3
#define HH 720
#define WW 1280
#define THREADS 256
#define EPSV 1e-6f
#define LOG2_WARP 0.49978214f   // log2(1.414)

// ---- sortable-uint encoding of float (monotonic): min in u32 == min in f32 ----
__device__ __forceinline__ unsigned f2s(float f) {
    unsigned b = __float_as_uint(f);
    return (b & 0x80000000u) ? ~b : (b | 0x80000000u);
}
__device__ __forceinline__ float s2f(unsigned u) {
    unsigned b = (u & 0x80000000u) ? (u ^ 0x80000000u) : ~u;
    return __uint_as_float(b);
}

// Async DMA: 16 bytes global -> LDS per enabled lane, tracked by ASYNCcnt.
__device__ __forceinline__ void async_copy_b128(const float* lds_dst, const float* gsrc) {
    const unsigned lds = (unsigned)(uintptr_t)lds_dst;                 // low 32b = LDS offset
    const unsigned long long ga = (unsigned long long)(uintptr_t)gsrc;
    asm volatile("global_load_async_to_lds_b128 %0, %1, off"
                 :: "v"(lds), "v"(ga) : "memory");
}
__device__ __forceinline__ void wait_asynccnt0() {
    asm volatile("s_wait_asynccnt 0x0" ::: "memory");
}

__global__ void fw_init_min(unsigned* __restrict__ ws) {
    ws[0] = f2s(FLT_MAX);
}

// Global min of disp: grid-stride float4, LDS tree, one global atomicMin per block.
__global__ __launch_bounds__(THREADS) void fw_min_reduce(const float* __restrict__ disp,
                                                         unsigned* __restrict__ ws, int n4) {
    const float4* d4 = (const float4*)disp;
    float m = FLT_MAX;
    for (int i = blockIdx.x * THREADS + threadIdx.x; i < n4; i += gridDim.x * THREADS) {
        float4 v = d4[i];
        m = fminf(m, fminf(fminf(v.x, v.y), fminf(v.z, v.w)));
    }
    __shared__ unsigned s[THREADS];
    s[threadIdx.x] = f2s(m);
    __syncthreads();
    for (int off = THREADS / 2; off > 0; off >>= 1) {
        if (threadIdx.x < off) s[threadIdx.x] = min(s[threadIdx.x], s[threadIdx.x + off]);
        __syncthreads();
    }
    if (threadIdx.x == 0) atomicMin(ws, s[0]);
}

// Scatter one pixel's two horizontal taps into the row accumulator (5 channels).
__device__ __forceinline__ void splat1(float* __restrict__ acc, int c, float d, float dmin,
                                       float p0, float p1, float p2) {
    const float wm = exp2f((d - dmin) * LOG2_WARP);   // 1.414^(d - dmin)
    const float x  = (float)c - d;                    // target x (flow_x = -disp)
    const float xf = floorf(x);
    const int   xi = (int)xf;
    const float w1 = x - xf;
    const float w0 = 1.0f - w1;
    const float a0 = p0 * wm, a1 = p1 * wm, a2 = p2 * wm;
    if (xi >= 0 && xi < WW) {                         // left tap, weight w0 (wy0 == 1)
        atomicAdd(acc + 0 * WW + xi, a0 * w0);
        atomicAdd(acc + 1 * WW + xi, a1 * w0);
        atomicAdd(acc + 2 * WW + xi, a2 * w0);
        atomicAdd(acc + 3 * WW + xi, wm * w0);        // wmap splat -> mask
        atomicAdd(acc + 4 * WW + xi, w0);             // ones splat -> occ
    }
    const int xj = xi + 1;
    if (xj >= 0 && xj < WW) {                         // right tap, weight w1
        atomicAdd(acc + 0 * WW + xj, a0 * w1);
        atomicAdd(acc + 1 * WW + xj, a1 * w1);
        atomicAdd(acc + 2 * WW + xj, a2 * w1);
        atomicAdd(acc + 3 * WW + xj, wm * w1);
        atomicAdd(acc + 4 * WW + xj, w1);
    }
}

// One block per (batch,row). disp < 40 => all scatter targets stay inside this row,
// so the whole splat resolves in LDS with ds_add_f32; no global atomics.
// Input rows are DMA'd into LDS with async b128 copies overlapped with LDS zeroing.
__global__ __launch_bounds__(THREADS) void fw_splat(const float* __restrict__ im,
                                                    const float* __restrict__ disp,
                                                    const unsigned* __restrict__ ws,
                                                    float* __restrict__ res,
                                                    float* __restrict__ occ) {
    __shared__ __align__(16) float acc[5 * WW];       // 25.6 KB accumulator
    __shared__ __align__(16) float stage[4 * WW];     // 20.5 KB input stage: disp,ch0,ch1,ch2
    const int row = blockIdx.x;                       // 0 .. BB*HH-1
    const int b   = row / HH;
    const int y   = row - b * HH;
    const int tid = threadIdx.x;

    const float* rows[4];
    rows[0] = disp + (size_t)row * WW;
    rows[1] = im + ((size_t)(b * CC + 0) * HH + y) * WW;
    rows[2] = im + ((size_t)(b * CC + 1) * HH + y) * WW;
    rows[3] = im + ((size_t)(b * CC + 2) * HH + y) * WW;

    // Kick off async global->LDS copies of the 4 input rows (ASYNCcnt).
#pragma unroll
    for (int r = 0; r < 4; ++r)
        for (int v = tid; v < WW / 4; v += THREADS)
            async_copy_b128(stage + r * WW + v * 4, rows[r] + v * 4);

    // Overlap: zero the accumulator with b128 LDS stores while the DMA runs.
    float4* accv = (float4*)acc;
    const float4 z = make_float4(0.f, 0.f, 0.f, 0.f);
    for (int i = tid; i < (5 * WW) / 4; i += THREADS) accv[i] = z;

    const float dmin = s2f(ws[0]);                    // uniform scalar load

    wait_asynccnt0();                                 // own staged chunks are consumed below
    __syncthreads();                                  // acc zeroed block-wide

    // Scatter pass entirely out of LDS: ds_load_b128 + 2 taps x 5 ds_add_f32 per pixel.
    const float4* sd = (const float4*)(stage + 0 * WW);
    const float4* s0 = (const float4*)(stage + 1 * WW);
    const float4* s1 = (const float4*)(stage + 2 * WW);
    const float4* s2 = (const float4*)(stage + 3 * WW);
    for (int v = tid; v < WW / 4; v += THREADS) {
        const float4 dd = sd[v];
        const float4 a  = s0[v];
        const float4 c4 = s1[v];
        const float4 e4 = s2[v];
        const int base = v * 4;
        splat1(acc, base + 0, dd.x, dmin, a.x, c4.x, e4.x);
        splat1(acc, base + 1, dd.y, dmin, a.y, c4.y, e4.y);
        splat1(acc, base + 2, dd.z, dmin, a.z, c4.z, e4.z);
        splat1(acc, base + 3, dd.w, dmin, a.w, c4.w, e4.w);
    }
    __syncthreads();

    // Normalize + occlusion, b128 stores.
    float4* r0 = (float4*)(res + ((size_t)(b * CC + 0) * HH + y) * WW);
    float4* r1 = (float4*)(res + ((size_t)(b * CC + 1) * HH + y) * WW);
    float4* r2 = (float4*)(res + ((size_t)(b * CC + 2) * HH + y) * WW);
    float4* ov = (float4*)(occ + (size_t)row * WW);
    const float4* a0 = (const float4*)(acc + 0 * WW);
    const float4* a1 = (const float4*)(acc + 1 * WW);
    const float4* a2 = (const float4*)(acc + 2 * WW);
    const float4* am = (const float4*)(acc + 3 * WW);
    const float4* ao = (const float4*)(acc + 4 * WW);
    for (int v = tid; v < WW / 4; v += THREADS) {
        const float4 m  = am[v];
        const float4 o  = ao[v];
        const float4 x0 = a0[v];
        const float4 x1 = a1[v];
        const float4 x2 = a2[v];
        const float ix = 1.0f / fmaxf(m.x, EPSV);
        const float iy = 1.0f / fmaxf(m.y, EPSV);
        const float iz = 1.0f / fmaxf(m.z, EPSV);
        const float iw = 1.0f / fmaxf(m.w, EPSV);
        r0[v] = make_float4(x0.x * ix, x0.y * iy, x0.z * iz, x0.w * iw);
        r1[v] = make_float4(x1.x * ix, x1.y * iy, x1.z * iz, x1.w * iw);
        r2[v] = make_float4(x2.x * ix, x2.y * iy, x2.z * iz, x2.w * iw);
        ov[v] = make_float4(1.0f - fminf(fmaxf(o.x, 0.0f), 1.0f),
                            1.0f - fminf(fmaxf(o.y, 0.0f), 1.0f),
                            1.0f - fminf(fmaxf(o.z, 0.0f), 1.0f),
                            1.0f - fminf(fmaxf(o.w, 0.0f), 1.0f));
    }
}

extern "C" void kernel_launch(void* const* d_in, const int* in_sizes, int n_in,
                              void* d_out, int out_size, void* d_ws, size_t ws_size,
                              hipStream_t stream) {
    (void)in_sizes; (void)n_in; (void)out_size; (void)ws_size;
    const float* im   = (const float*)d_in[0];
    const float* disp = (const float*)d_in[1];
    float* res = (float*)d_out;                               // [8,3,720,1280]
    float* occ = res + (size_t)BB * CC * HH * WW;             // [8,1,720,1280]
    unsigned* ws = (unsigned*)d_ws;

    fw_init_min<<<1, 1, 0, stream>>>(ws);
    const int n4 = (BB * HH * WW) / 4;
    fw_min_reduce<<<1440, THREADS, 0, stream>>>(disp, ws, n4);
    fw_splat<<<BB * HH, THREADS, 0, stream>>>(im, disp, ws, res, occ);
}